// MMLoss_29180007809317
// MI455X (gfx1250) — compile-verified
//
#include <hip/hip_runtime.h>
#include <math.h>

#define B_   4
#define C_   128
#define H_   192
#define W_   192
#define HW_  (H_*W_)
#define NS   4096
#define CP   132          // padded LDS row stride (floats): 528B rows -> 16B aligned, bank-skewed
#define PI_F 3.14159265358979323846f

typedef float v2f __attribute__((ext_vector_type(2)));
typedef float v8f __attribute__((ext_vector_type(8)));

// ---- order-preserving float <-> uint encoding for atomicMax ----------------
__device__ __forceinline__ unsigned encf(float f) {
  unsigned u = __float_as_uint(f);
  return (u & 0x80000000u) ? ~u : (u | 0x80000000u);
}
__device__ __forceinline__ float decf(unsigned k) {
  unsigned u = (k & 0x80000000u) ? (k ^ 0x80000000u) : ~k;
  return __uint_as_float(u);
}
__device__ __forceinline__ float clampv(float x) {
  return fminf(fmaxf(x, -1.0f + 1e-5f), 1.0f - 1e-5f);
}

// ---- 1) AvgPool2d(3, stride 1, pad 1, count_include_pad) -------------------
__global__ void avgpool3_k(const float* __restrict__ in, float* __restrict__ out) {
  int g = blockIdx.x * blockDim.x + threadIdx.x;
  if (g >= B_ * HW_) return;
  int x = g % W_, y = (g / W_) % H_, b = g / HW_;
  const float* p = in + (size_t)b * HW_;
  float s = 0.f;
  for (int dy = -1; dy <= 1; ++dy) {
    int yy = y + dy; if (yy < 0 || yy >= H_) continue;
    for (int dx = -1; dx <= 1; ++dx) {
      int xx = x + dx; if (xx < 0 || xx >= W_) continue;
      s += p[yy * W_ + xx];
    }
  }
  out[g] = s * (1.f / 9.f);
}

// ---- 2) exact top-4096 select: radix binary search + ordered compaction ----
__global__ __launch_bounds__(1024) void topk_select_k(const float* __restrict__ rv,
                                                      const float* __restrict__ gm,
                                                      int* __restrict__ idxOut) {
  const int b = blockIdx.x, tid = threadIdx.x;
  const float* R = rv + (size_t)b * HW_;
  const float* G = gm + (size_t)b * HW_;
  __shared__ unsigned s_cnt;
  __shared__ unsigned s_scan[1024];
  __shared__ unsigned s_base;

  // largest threshold T with count(key >= T) >= NS  (keys distinct w.p. 1)
  unsigned long long lo = 0ull, hi = 0xFFFFFFFFull;
  for (int it = 0; it < 33; ++it) {
    if (lo >= hi) break;                      // uniform across block
    unsigned long long mid = lo + ((hi - lo + 1) >> 1);
    __syncthreads();
    if (tid == 0) s_cnt = 0;
    __syncthreads();
    unsigned local = 0;
    for (int e = tid; e < HW_; e += 1024)
      if ((unsigned long long)encf(R[e] * G[e]) >= mid) local++;
    atomicAdd(&s_cnt, local);
    __syncthreads();
    unsigned c = s_cnt;
    if (c >= NS) lo = mid; else hi = mid - 1;
  }
  const unsigned T = (unsigned)lo;
  __syncthreads();
  if (tid == 0) s_base = 0;

  for (int chunk = 0; chunk < HW_; chunk += 1024) {   // HW_ % 1024 == 0
    int e = chunk + tid;
    unsigned flag = (encf(R[e] * G[e]) >= T) ? 1u : 0u;
    __syncthreads();
    s_scan[tid] = flag;
    __syncthreads();
    for (int off = 1; off < 1024; off <<= 1) {        // Hillis-Steele inclusive scan
      unsigned v = (tid >= off) ? s_scan[tid - off] : 0u;
      __syncthreads();
      s_scan[tid] += v;
      __syncthreads();
    }
    unsigned incl = s_scan[tid];
    unsigned total = s_scan[1023];
    unsigned base = s_base;
    if (flag) {
      unsigned p = base + incl - 1u;
      if (p < NS) idxOut[b * NS + p] = e;             // deterministic, index-ordered
    }
    __syncthreads();
    if (tid == 0) s_base = base + total;
  }
}

// ---- 3) gather descriptors / coords / pooled-score products ----------------
__global__ __launch_bounds__(128) void gather_k(const float* __restrict__ f1,
                                                const float* __restrict__ f2,
                                                const float* __restrict__ s1p,
                                                const float* __restrict__ s2p,
                                                const int* __restrict__ idx,
                                                float* __restrict__ f1s, float* __restrict__ f2s,
                                                float* __restrict__ px, float* __restrict__ py,
                                                float* __restrict__ score) {
  const int n = blockIdx.x, b = blockIdx.y, c = threadIdx.x;
  const int id = idx[b * NS + n];
  const size_t srcb = (size_t)b * C_ * HW_;
  f1s[((size_t)(b * NS + n)) * C_ + c] = f1[srcb + (size_t)c * HW_ + id];
  f2s[((size_t)(b * NS + n)) * C_ + c] = f2[srcb + (size_t)c * HW_ + id];
  if (c == 0) {
    px[b * NS + n] = (float)(id % W_);
    py[b * NS + n] = (float)(id / W_);
    score[b * NS + n] = s1p[b * HW_ + id] * s2p[b * HW_ + id];
  }
}

__global__ void zero_u32_k(unsigned* p, int n) {
  int g = blockIdx.x * blockDim.x + threadIdx.x;
  if (g < n) p[g] = 0u;
}

// ---- 4) WMMA f32 tile GEMM + mask + col/row max reductions -----------------
// grid: (NS/64, NS/64, B_*3); z = b*3 + mat;  mat 0: f1*f2^T, 1: f1*f1^T, 2: f2*f2^T
// 128 threads = 4 waves; each wave owns a 32x32 register tile (4 accumulators),
// so each K-step is 4 ds_load_b64 : 4 v_wmma (A/B fragments reused twice each).
__global__ __launch_bounds__(128) void hard_tiles_k(const float* __restrict__ f1s,
                                                    const float* __restrict__ f2s,
                                                    const float* __restrict__ px,
                                                    const float* __restrict__ py,
                                                    unsigned* __restrict__ posE,
                                                    unsigned* __restrict__ negnE,
                                                    unsigned* __restrict__ negmE,
                                                    unsigned* __restrict__ negkE,
                                                    unsigned* __restrict__ negjE) {
  __shared__ float As[64][CP];
  __shared__ float Bs[64][CP];
  __shared__ float pxi[64], pyi[64], pxj[64], pyj[64];

  const int mat = blockIdx.z % 3;
  const int b   = blockIdx.z / 3;
  const int i0  = blockIdx.x * 64;
  const int j0  = blockIdx.y * 64;

  const float* Aab = (mat == 2) ? f2s : f1s;
  const float* Bab = (mat == 0) ? f2s : Aab;
  unsigned* colOut = (mat == 0) ? negnE : ((mat == 1) ? negkE : negjE);
  const float* Ag = Aab + (size_t)b * NS * C_;
  const float* Bg = Bab + (size_t)b * NS * C_;

  const int tid = threadIdx.x;
  { // A and B tiles 64x128 each (row-major); rows are 16B aligned (CP%4==0)
    int r = tid >> 1, cq = (tid & 1) * 64;
    const float4* srcA = (const float4*)(Ag + (size_t)(i0 + r) * C_ + cq);
    const float4* srcB = (const float4*)(Bg + (size_t)(j0 + r) * C_ + cq);
    #pragma unroll
    for (int q = 0; q < 16; ++q) *(float4*)&As[r][cq + q * 4] = srcA[q];
    #pragma unroll
    for (int q = 0; q < 16; ++q) *(float4*)&Bs[r][cq + q * 4] = srcB[q];
  }
  if (tid < 64) {
    pxi[tid] = px[b * NS + i0 + tid]; pyi[tid] = py[b * NS + i0 + tid];
    pxj[tid] = px[b * NS + j0 + tid]; pyj[tid] = py[b * NS + j0 + tid];
  }
  __syncthreads();

  const int lane  = tid & 31;
  const int wv    = tid >> 5;
  const int rbase = (wv >> 1) * 32;      // wave's 32-row band within the 64-row tile
  const int cbase = (wv & 1) * 32;       // wave's 32-col band within the 64-col tile
  const int frow  = lane & 15;
  const int koff  = (lane >> 4) << 1;    // K pair select per ISA A/B layout
  const int hi8   = (lane >> 4) * 8;     // C layout: M = v + 8*(lane>=16)

  v8f acc[2][2];
  #pragma unroll
  for (int rt = 0; rt < 2; ++rt)
    #pragma unroll
    for (int ct = 0; ct < 2; ++ct)
      acc[rt][ct] = (v8f){0.f,0.f,0.f,0.f,0.f,0.f,0.f,0.f};

  #pragma unroll 4
  for (int k = 0; k < C_; k += 4) {
    v2f a0 = *(const v2f*)&As[rbase + frow][k + koff];
    v2f a1 = *(const v2f*)&As[rbase + 16 + frow][k + koff];
    v2f b0 = *(const v2f*)&Bs[cbase + frow][k + koff];
    v2f b1 = *(const v2f*)&Bs[cbase + 16 + frow][k + koff];
    acc[0][0] = __builtin_amdgcn_wmma_f32_16x16x4_f32(false, a0, false, b0, (short)0, acc[0][0], false, false);
    acc[0][1] = __builtin_amdgcn_wmma_f32_16x16x4_f32(false, a0, false, b1, (short)0, acc[0][1], false, false);
    acc[1][0] = __builtin_amdgcn_wmma_f32_16x16x4_f32(false, a1, false, b0, (short)0, acc[1][0], false, false);
    acc[1][1] = __builtin_amdgcn_wmma_f32_16x16x4_f32(false, a1, false, b1, (short)0, acc[1][1], false, false);
  }

  // ---- epilogue: masked values + reductions ----
  const int   nloc = lane & 15;
  float pxc[2], pyc[2];
  pxc[0] = pxj[cbase + nloc];      pyc[0] = pyj[cbase + nloc];
  pxc[1] = pxj[cbase + 16 + nloc]; pyc[1] = pyj[cbase + 16 + nloc];

  float colneg[2] = {-1e30f, -1e30f};
  float colpos[2] = {-1e30f, -1e30f};
  float rowmax[2][8];

  #pragma unroll
  for (int rt = 0; rt < 2; ++rt) {
    #pragma unroll
    for (int v = 0; v < 8; ++v) {
      const int   rl  = rbase + rt * 16 + v + hi8;
      const float pxr = pxi[rl], pyr = pyi[rl];
      float rm = -1e30f;
      #pragma unroll
      for (int ct = 0; ct < 2; ++ct) {
        float s  = acc[rt][ct][v];
        float dx = pxr - pxc[ct], dy = pyr - pyc[ct];
        float d2 = dx * dx + dy * dy;
        float nv = s - ((s > 0.9f) ? 10.f : 0.f) - ((d2 < 49.f) ? 10.f : 0.f);
        colneg[ct] = fmaxf(colneg[ct], nv);
        rm = fmaxf(rm, nv);
        colpos[ct] = fmaxf(colpos[ct], s - ((d2 < 9.f) ? 0.f : 10.f));
      }
      rowmax[rt][v] = rm;
    }
  }

  // column maxima: combine the two M-halves (lane ^ 16 has same N)
  #pragma unroll
  for (int ct = 0; ct < 2; ++ct) {
    colneg[ct] = fmaxf(colneg[ct], __shfl_xor(colneg[ct], 16, 32));
    colpos[ct] = fmaxf(colpos[ct], __shfl_xor(colpos[ct], 16, 32));
  }
  if (lane < 16) {
    atomicMax(&colOut[b * NS + j0 + cbase + lane],      encf(colneg[0]));
    atomicMax(&colOut[b * NS + j0 + cbase + 16 + lane], encf(colneg[1]));
    if (mat == 0) {
      atomicMax(&posE[b * NS + j0 + cbase + lane],      encf(colpos[0]));
      atomicMax(&posE[b * NS + j0 + cbase + 16 + lane], encf(colpos[1]));
    }
  }
  // row maxima (only needed for the cross-similarity matrix)
  if (mat == 0) {
    #pragma unroll
    for (int rt = 0; rt < 2; ++rt) {
      #pragma unroll
      for (int v = 0; v < 8; ++v) {
        float rv = rowmax[rt][v];
        rv = fmaxf(rv, __shfl_xor(rv, 1, 32));
        rv = fmaxf(rv, __shfl_xor(rv, 2, 32));
        rv = fmaxf(rv, __shfl_xor(rv, 4, 32));
        rv = fmaxf(rv, __shfl_xor(rv, 8, 32));
        if ((lane & 15) == 0)
          atomicMax(&negmE[b * NS + i0 + rbase + rt * 16 + v + hi8], encf(rv));
      }
    }
  }
}

// ---- 5) per-batch arccos epilogue + reductions -----------------------------
__global__ __launch_bounds__(256) void finalize_batch_k(const float* __restrict__ score,
                                                        const unsigned* __restrict__ posE,
                                                        const unsigned* __restrict__ negnE,
                                                        const unsigned* __restrict__ negmE,
                                                        const unsigned* __restrict__ negkE,
                                                        const unsigned* __restrict__ negjE,
                                                        float* __restrict__ Lb,
                                                        float* __restrict__ Sb) {
  const int b = blockIdx.x, tid = threadIdx.x;
  float aL = 0.f, aS = 0.f;
  for (int k = tid; k < NS; k += 256) {
    const int o = b * NS + k;
    float sc  = score[o];
    aS += sc;
    float pos = clampv(decf(posE[o]));
    float ncr = clampv(fmaxf(decf(negnE[o]), decf(negmE[o])));
    float nk  = clampv(decf(negkE[o]));
    float nj  = clampv(decf(negjE[o]));
    float t, m;
    t = PI_F - acosf(nk);  m  = t * t * (1.f / 3.f);
    t = PI_F - acosf(nj);  m += t * t * (1.f / 3.f);
    t = PI_F - acosf(ncr); m += t * t * (1.f / 3.f);
    t = acosf(pos);        m += t * t;
    aL += sc * m * m;
  }
  __shared__ float rL[256], rS[256];
  rL[tid] = aL; rS[tid] = aS;
  __syncthreads();
  for (int s = 128; s > 0; s >>= 1) {
    if (tid < s) { rL[tid] += rL[tid + s]; rS[tid] += rS[tid + s]; }
    __syncthreads();
  }
  if (tid == 0) { Lb[b] = rL[0]; Sb[b] = rS[0]; }
}

// ---- 6) sequential rsum recurrence -> scalar -------------------------------
__global__ void final_scalar_k(const float* __restrict__ Lb, const float* __restrict__ Sb,
                               float* __restrict__ out) {
  float rsum = 1000.f, loss = 0.f;
  for (int b = 0; b < B_; ++b) {
    loss += Lb[b] / (rsum + 1e-5f);
    rsum  = 0.99f * rsum + 0.01f * Sb[b];
  }
  out[0] = loss / (float)B_;
}

// ---------------------------------------------------------------------------
extern "C" void kernel_launch(void* const* d_in, const int* in_sizes, int n_in,
                              void* d_out, int out_size, void* d_ws, size_t ws_size,
                              hipStream_t stream) {
  const float* feat1  = (const float*)d_in[0];
  const float* feat2  = (const float*)d_in[1];
  const float* score1 = (const float*)d_in[2];
  const float* score2 = (const float*)d_in[3];
  const float* gmask  = (const float*)d_in[4];
  const float* randv  = (const float*)d_in[5];
  float* out = (float*)d_out;

  char* w = (char*)d_ws;
  auto alloc = [&](size_t bytes) -> void* {
    void* p = (void*)w;
    w += (bytes + 255) & ~(size_t)255;
    return p;
  };
  float* s1p   = (float*)alloc(sizeof(float) * B_ * HW_);
  float* s2p   = (float*)alloc(sizeof(float) * B_ * HW_);
  int*   idx   = (int*)  alloc(sizeof(int)   * B_ * NS);
  float* f1s   = (float*)alloc(sizeof(float) * (size_t)B_ * NS * C_);
  float* f2s   = (float*)alloc(sizeof(float) * (size_t)B_ * NS * C_);
  float* px    = (float*)alloc(sizeof(float) * B_ * NS);
  float* py    = (float*)alloc(sizeof(float) * B_ * NS);
  float* score = (float*)alloc(sizeof(float) * B_ * NS);
  unsigned* encBase = (unsigned*)alloc(sizeof(unsigned) * 5 * B_ * NS);
  unsigned* posE  = encBase;
  unsigned* negnE = encBase + 1 * B_ * NS;
  unsigned* negmE = encBase + 2 * B_ * NS;
  unsigned* negkE = encBase + 3 * B_ * NS;
  unsigned* negjE = encBase + 4 * B_ * NS;
  float* Lb = (float*)alloc(sizeof(float) * B_);
  float* Sb = (float*)alloc(sizeof(float) * B_);

  avgpool3_k<<<(B_ * HW_ + 255) / 256, 256, 0, stream>>>(score1, s1p);
  avgpool3_k<<<(B_ * HW_ + 255) / 256, 256, 0, stream>>>(score2, s2p);
  topk_select_k<<<B_, 1024, 0, stream>>>(randv, gmask, idx);
  gather_k<<<dim3(NS, B_), 128, 0, stream>>>(feat1, feat2, s1p, s2p, idx,
                                             f1s, f2s, px, py, score);
  zero_u32_k<<<(5 * B_ * NS + 255) / 256, 256, 0, stream>>>(encBase, 5 * B_ * NS);
  hard_tiles_k<<<dim3(NS / 64, NS / 64, B_ * 3), 128, 0, stream>>>(
      f1s, f2s, px, py, posE, negnE, negmE, negkE, negjE);
  finalize_batch_k<<<B_, 256, 0, stream>>>(score, posE, negnE, negmE, negkE, negjE, Lb, Sb);
  final_scalar_k<<<1, 1, 0, stream>>>(Lb, Sb, out);
}